// CGProduct_22488448762743
// MI455X (gfx1250) — compile-verified
//
#include <hip/hip_runtime.h>

typedef __attribute__((ext_vector_type(2))) float v2f;
typedef __attribute__((ext_vector_type(8))) float v8f;

// ---------------- compile-time Clebsch-Gordan coefficients ----------------
constexpr double cfact(int n) { double r = 1.0; for (int i = 2; i <= n; ++i) r *= (double)i; return r; }
constexpr double csqrt(double x) {
  if (x <= 0.0) return 0.0;
  double g = (x > 1.0) ? x : 1.0;
  for (int i = 0; i < 60; ++i) g = 0.5 * (g + x / g);
  return g;
}
constexpr double cg_scalar(int j1, int m1, int j2, int m2, int j, int m) {
  if (m1 + m2 != m) return 0.0;
  double pref = csqrt((double)(2 * j + 1) * cfact(j + j1 - j2) * cfact(j - j1 + j2) *
                      cfact(j1 + j2 - j) / cfact(j1 + j2 + j + 1));
  pref *= csqrt(cfact(j + m) * cfact(j - m) * cfact(j1 + m1) * cfact(j1 - m1) *
                cfact(j2 + m2) * cfact(j2 - m2));
  double s = 0.0;
  for (int k = 0; k <= j1 + j2 - j; ++k) {
    int d1 = j1 + j2 - j - k, d2 = j1 - m1 - k, d3 = j2 + m2 - k;
    int d4 = j - j2 + m1 + k, d5 = j - j1 - m2 + k;
    if (d1 < 0 || d2 < 0 || d3 < 0 || d4 < 0 || d5 < 0) continue;
    double denom = cfact(k) * cfact(d1) * cfact(d2) * cfact(d3) * cfact(d4) * cfact(d5);
    s += ((k & 1) ? -1.0 : 1.0) / denom;
  }
  return pref * s;
}

constexpr int loff(int l) { return (l == 0) ? 0 : (l == 1) ? 1 : (l == 2) ? 4 : 9; }

constexpr float cgcoef(int L1, int L2, int L, int M, int r) {
  if (r > 2 * L1) return 0.0f;
  int mm1 = r - L1, mm = M - L, mm2 = mm - mm1;
  if (mm2 < -L2 || mm2 > L2) return 0.0f;
  return (float)cg_scalar(L1, mm1, L2, mm2, L, mm);
}
constexpr int cgbidx(int L1, int L2, int L, int M, int r) {
  if (r > 2 * L1) return 0;
  int mm2 = (M - L) - (r - L1);
  if (mm2 < -L2 || mm2 > L2) return 0;
  return mm2 + L2;
}

// ---------------- opaque half-select: exactly one v_perm_b32 ----------------
// sel = 0x07060504 (take hi) for lanes 16..31, 0x03020100 (take lo) for lanes 0..15.
// Using v_perm with a runtime selector blocks InstCombine's
// select(load)/indexed-extract canonicalization that caused cndmask waterfalls.
__device__ __forceinline__ float hsel(unsigned sel, float hi, float lo) {
  return __builtin_bit_cast(
      float, __builtin_amdgcn_perm(__builtin_bit_cast(unsigned, hi),
                                   __builtin_bit_cast(unsigned, lo), sel));
}

// A operand component: rows RLO (lanes 0-15) / RHI (lanes 16-31), zero past K1.
template <int L1, int RLO, int RHI>
__device__ __forceinline__ float aop(unsigned sel, const float* Ar) {
  if constexpr (RHI <= 2 * L1) return hsel(sel, Ar[loff(L1) + RHI], Ar[loff(L1) + RLO]);
  else if constexpr (RLO <= 2 * L1) return hsel(sel, 0.0f, Ar[loff(L1) + RLO]);
  else return 0.0f;
}

// Btilde operand component: CG[M,r,m2*] * B[m2*, q], row r per half.
template <int L1, int L2, int L, int M, int RLO, int RHI>
__device__ __forceinline__ float bop(unsigned sel, const float* Br) {
  constexpr float CLO = cgcoef(L1, L2, L, M, RLO);
  constexpr float CHI = cgcoef(L1, L2, L, M, RHI);
  if constexpr (CLO == 0.0f && CHI == 0.0f) {
    return 0.0f;
  } else if constexpr (CHI == 0.0f) {
    constexpr int ILO = loff(L2) + cgbidx(L1, L2, L, M, RLO);
    return hsel(sel, 0.0f, CLO * Br[ILO]);
  } else if constexpr (CLO == 0.0f) {
    constexpr int IHI = loff(L2) + cgbidx(L1, L2, L, M, RHI);
    return hsel(sel, CHI * Br[IHI], 0.0f);
  } else {
    constexpr int ILO = loff(L2) + cgbidx(L1, L2, L, M, RLO);
    constexpr int IHI = loff(L2) + cgbidx(L1, L2, L, M, RHI);
    return hsel(sel, CHI * Br[IHI], CLO * Br[ILO]);
  }
}

template <int L1, int L2, int L, int M, int KT>
__device__ __forceinline__ v8f ktile(unsigned sel, v8f acc, const float* Ar, const float* Br) {
  v2f a, b;
  // k = 4*KT + 2*half + v
  a.x = aop<L1, 4 * KT + 0, 4 * KT + 2>(sel, Ar);
  a.y = aop<L1, 4 * KT + 1, 4 * KT + 3>(sel, Ar);
  b.x = bop<L1, L2, L, M, 4 * KT + 0, 4 * KT + 2>(sel, Br);
  b.y = bop<L1, L2, L, M, 4 * KT + 1, 4 * KT + 3>(sel, Br);
  return __builtin_amdgcn_wmma_f32_16x16x4_f32(false, a, false, b, (short)0, acc, false, false);
}

// One (l1,l2)->l pair; all output offsets folded to compile-time store immediates.
template <int L1, int L2, int L, int NP, int POFF, int BASE, int M = 0>
__device__ __forceinline__ void do_pair(unsigned sel, const float* Ar, const float* Br,
                                        float* pbase) {
  if constexpr (M < 2 * L + 1) {
    v8f acc = {};
    acc = ktile<L1, L2, L, M, 0>(sel, acc, Ar, Br);
    if constexpr (2 * L1 + 1 > 4) acc = ktile<L1, L2, L, M, 1>(sel, acc, Ar, Br);
    constexpr int OFF = BASE + M * (NP * 256) + POFF * 256;
#pragma unroll
    for (int v = 0; v < 8; ++v) pbase[OFF + v * 16] = acc[v];
    do_pair<L1, L2, L, NP, POFF, BASE, M + 1>(sel, Ar, Br, pbase);
  }
}

// Per-batch output layout: l-bases 0 / 1024 / 7936 / 22016, total 39936 floats.
#define OUT_PER_B 39936

__global__ __launch_bounds__(256) void cg_product_wmma_kernel(
    const float* __restrict__ A0, const float* __restrict__ A1,
    const float* __restrict__ A2, const float* __restrict__ A3,
    const float* __restrict__ B0, const float* __restrict__ B1,
    const float* __restrict__ B2, const float* __restrict__ B3,
    float* __restrict__ out, int batch) {
  int wid = (int)((blockIdx.x * blockDim.x + threadIdx.x) >> 5);
  if (wid >= batch) return;  // uniform per wave; EXEC all-ones inside
  const int lane = threadIdx.x & 31;
  const int q = lane & 15;
  const unsigned sel = (lane & 16) ? 0x07060504u : 0x03020100u;  // hi/lo byte-selector
  const size_t b = (size_t)wid;

  // Per-lane column q of every irrep (lanes 16-31 mirror 0-15).
  float Ar[16], Br[16];
  Ar[0] = A0[b * 16 + q];
  Br[0] = B0[b * 16 + q];
#pragma unroll
  for (int m = 0; m < 3; ++m) { Ar[1 + m] = A1[(b * 3 + m) * 16 + q]; Br[1 + m] = B1[(b * 3 + m) * 16 + q]; }
#pragma unroll
  for (int m = 0; m < 5; ++m) { Ar[4 + m] = A2[(b * 5 + m) * 16 + q]; Br[4 + m] = B2[(b * 5 + m) * 16 + q]; }
#pragma unroll
  for (int m = 0; m < 7; ++m) { Ar[9 + m] = A3[(b * 7 + m) * 16 + q]; Br[9 + m] = B3[(b * 7 + m) * 16 + q]; }

  // Single store base; every store offset is a compile-time immediate (<160KB < 8MB).
  const int rq = q + ((lane & 16) << 3);  // q + half*128
  float* pbase = out + b * (size_t)OUT_PER_B + rq;

  // ---- l = 0 (NP=4, BASE=0) ----
  do_pair<0, 0, 0, 4, 0, 0>(sel, Ar, Br, pbase);
  do_pair<1, 1, 0, 4, 1, 0>(sel, Ar, Br, pbase);
  do_pair<2, 2, 0, 4, 2, 0>(sel, Ar, Br, pbase);
  do_pair<3, 3, 0, 4, 3, 0>(sel, Ar, Br, pbase);
  // ---- l = 1 (NP=9, BASE=1024) ----
  do_pair<0, 1, 1, 9, 0, 1024>(sel, Ar, Br, pbase);
  do_pair<1, 0, 1, 9, 1, 1024>(sel, Ar, Br, pbase);
  do_pair<1, 1, 1, 9, 2, 1024>(sel, Ar, Br, pbase);
  do_pair<1, 2, 1, 9, 3, 1024>(sel, Ar, Br, pbase);
  do_pair<2, 1, 1, 9, 4, 1024>(sel, Ar, Br, pbase);
  do_pair<2, 2, 1, 9, 5, 1024>(sel, Ar, Br, pbase);
  do_pair<2, 3, 1, 9, 6, 1024>(sel, Ar, Br, pbase);
  do_pair<3, 2, 1, 9, 7, 1024>(sel, Ar, Br, pbase);
  do_pair<3, 3, 1, 9, 8, 1024>(sel, Ar, Br, pbase);
  // ---- l = 2 (NP=11, BASE=7936) ----
  do_pair<0, 2, 2, 11, 0, 7936>(sel, Ar, Br, pbase);
  do_pair<1, 1, 2, 11, 1, 7936>(sel, Ar, Br, pbase);
  do_pair<1, 2, 2, 11, 2, 7936>(sel, Ar, Br, pbase);
  do_pair<1, 3, 2, 11, 3, 7936>(sel, Ar, Br, pbase);
  do_pair<2, 0, 2, 11, 4, 7936>(sel, Ar, Br, pbase);
  do_pair<2, 1, 2, 11, 5, 7936>(sel, Ar, Br, pbase);
  do_pair<2, 2, 2, 11, 6, 7936>(sel, Ar, Br, pbase);
  do_pair<2, 3, 2, 11, 7, 7936>(sel, Ar, Br, pbase);
  do_pair<3, 1, 2, 11, 8, 7936>(sel, Ar, Br, pbase);
  do_pair<3, 2, 2, 11, 9, 7936>(sel, Ar, Br, pbase);
  do_pair<3, 3, 2, 11, 10, 7936>(sel, Ar, Br, pbase);
  // ---- l = 3 (NP=10, BASE=22016) ----
  do_pair<0, 3, 3, 10, 0, 22016>(sel, Ar, Br, pbase);
  do_pair<1, 2, 3, 10, 1, 22016>(sel, Ar, Br, pbase);
  do_pair<1, 3, 3, 10, 2, 22016>(sel, Ar, Br, pbase);
  do_pair<2, 1, 3, 10, 3, 22016>(sel, Ar, Br, pbase);
  do_pair<2, 2, 3, 10, 4, 22016>(sel, Ar, Br, pbase);
  do_pair<2, 3, 3, 10, 5, 22016>(sel, Ar, Br, pbase);
  do_pair<3, 0, 3, 10, 6, 22016>(sel, Ar, Br, pbase);
  do_pair<3, 1, 3, 10, 7, 22016>(sel, Ar, Br, pbase);
  do_pair<3, 2, 3, 10, 8, 22016>(sel, Ar, Br, pbase);
  do_pair<3, 3, 3, 10, 9, 22016>(sel, Ar, Br, pbase);
}

extern "C" void kernel_launch(void* const* d_in, const int* in_sizes, int n_in,
                              void* d_out, int out_size, void* d_ws, size_t ws_size,
                              hipStream_t stream) {
  (void)n_in; (void)out_size; (void)d_ws; (void)ws_size;
  // setup_inputs() dict order: A0, B0, A1, B1, A2, B2, A3, B3
  const float* A0 = (const float*)d_in[0];
  const float* B0 = (const float*)d_in[1];
  const float* A1 = (const float*)d_in[2];
  const float* B1 = (const float*)d_in[3];
  const float* A2 = (const float*)d_in[4];
  const float* B2 = (const float*)d_in[5];
  const float* A3 = (const float*)d_in[6];
  const float* B3 = (const float*)d_in[7];
  float* out = (float*)d_out;

  const int batch = in_sizes[0] / 16;  // A0 is [batch, 1, 16]
  const int threads = 256;             // 8 waves per block
  const int blocks = (batch * 32 + threads - 1) / threads;
  cg_product_wmma_kernel<<<blocks, threads, 0, stream>>>(A0, A1, A2, A3, B0, B1, B2, B3,
                                                         out, batch);
}